// GNNEncoder_18279380812528
// MI455X (gfx1250) — compile-verified
//
#include <hip/hip_runtime.h>
#include <hip/hip_bf16.h>

#define NN 50000
#define EE 600000
#define HID 128
#define APAD 132   // LDS row stride (floats): 128 + 4 -> conflict-free b64 reads

typedef __attribute__((ext_vector_type(2))) float v2f;
typedef __attribute__((ext_vector_type(8))) float v8f;

// Non-returning global fp32 atomic add (lowers to global_atomic_add_f32, STOREcnt-tracked).
__device__ __forceinline__ void gatomic_add_f32(float* addr, float val) {
    asm volatile("global_atomic_add_f32 %0, %1, off" : : "v"(addr), "v"(val) : "memory");
}

// ---------------------------------------------------------------------------
// Zero a float range (grid-stride).
// ---------------------------------------------------------------------------
__global__ void zero_kernel(float* __restrict__ p, long long n) {
    long long i = (long long)blockIdx.x * blockDim.x + threadIdx.x;
    long long stride = (long long)gridDim.x * blockDim.x;
    for (; i < n; i += stride) p[i] = 0.0f;
}

// ---------------------------------------------------------------------------
// Edge scatter: one wave per edge, each lane handles 4 contiguous channels.
//   agg[dst] += feat[src];  cnt[dst] += 1 (lane 0, first layer only)
// feat rows are 512B; gathers/atomics stay in the 192MB L2 (x/agg/h resident).
// ---------------------------------------------------------------------------
__global__ void scatter_kernel(const float* __restrict__ feat,
                               const long long* __restrict__ ei,
                               float* __restrict__ agg,
                               float* cnt, int nedges) {
    int e = blockIdx.x * 8 + (threadIdx.x >> 5);
    if (e >= nedges) return;
    int lane = threadIdx.x & 31;
    long long s = ei[e];            // src node
    long long d = ei[nedges + e];   // dst node

    const float4 v = *(const float4*)(feat + s * HID + lane * 4);
    float* p = agg + d * HID + lane * 4;
    gatomic_add_f32(p + 0, v.x);
    gatomic_add_f32(p + 1, v.y);
    gatomic_add_f32(p + 2, v.z);
    gatomic_add_f32(p + 3, v.w);
    if (cnt != nullptr && lane == 0) gatomic_add_f32(cnt + d, 1.0f);
}

// ---------------------------------------------------------------------------
// Fused SAGE GEMM:
//   out = (agg/max(cnt,1)) @ Wl^T + bl + xin @ Wr^T   (optional ReLU)
//
// Block = 256 thr = 8 waves; block owns one 16-row stripe, wave w owns the
// 16-col tile w (8 x 16 = full 128 cols). Grid = 3125 row tiles (50000/16
// exact -> EXEC all-ones for WMMA).
//
// The shared 16x128 A tiles (mean-scaled agg, and x) are staged in LDS once
// per block (16.5 KB) instead of each wave re-reading them from L2 (8x
// de-duplication); the mean division happens once at staging time.
//
// Fragment layout (ISA 7.12.2, 32-bit):
//   A (16x4):  lane l -> M = l%16, K = kbase + 2*(l/16) + {vgpr 0,1}
//   B (4x16):  lane l -> N = l%16, K = kbase + 2*(l/16) + {vgpr 0,1}
//   C (16x16): lane l, vgpr j -> M = j + 8*(l/16), N = l%16
// LDS banks: addr/4 = idx*APAD + k0 = 132*idx + ... -> bank 4*idx + 2*half
// mod 64: all 32 lanes hit disjoint bank pairs for the b64 reads.
// ---------------------------------------------------------------------------
__global__ void sage_gemm_kernel(const float* __restrict__ agg,
                                 const float* __restrict__ cnt,
                                 const float* __restrict__ xin,
                                 const float* __restrict__ Wl,
                                 const float* __restrict__ bl,
                                 const float* __restrict__ Wr,
                                 float* __restrict__ out,
                                 int do_relu) {
    __shared__ float sMean[16][APAD];
    __shared__ float sX[16][APAD];

    const int lane = threadIdx.x & 31;
    const int wave = threadIdx.x >> 5;   // column tile 0..7
    const int half = lane >> 4;          // 0 or 1
    const int idx  = lane & 15;

    const int rowBase = blockIdx.x * 16;
    const int col     = wave * 16 + idx;     // B-col this lane feeds

    // ---- Stage A tiles into LDS (2048 floats per matrix = 512 float4) ----
    for (int i = threadIdx.x; i < 512; i += 256) {
        const int r  = i >> 5;          // row 0..15
        const int c4 = (i & 31) * 4;    // col 0,4,...,124
        const float inv = 1.0f / fmaxf(cnt[rowBase + r], 1.0f);
        float4 va = *(const float4*)(agg + (size_t)(rowBase + r) * HID + c4);
        va.x *= inv; va.y *= inv; va.z *= inv; va.w *= inv;
        *(float4*)(&sMean[r][c4]) = va;
        *(float4*)(&sX[r][c4]) =
            *(const float4*)(xin + (size_t)(rowBase + r) * HID + c4);
    }
    __syncthreads();

    const float* wlCol = Wl + (size_t)col * HID;   // Wl[n][k] == (Wl^T)[k][n]
    const float* wrCol = Wr + (size_t)col * HID;

    v8f acc = {};

#pragma unroll 8
    for (int kk = 0; kk < HID / 4; ++kk) {
        const int k0 = kk * 4 + half * 2;

        float2 tm = *(const float2*)(&sMean[idx][k0]);
        float2 tx = *(const float2*)(&sX[idx][k0]);
        float2 tl = *(const float2*)(wlCol + k0);
        float2 tr = *(const float2*)(wrCol + k0);

        v2f aM, aX, bL, bR;
        aM.x = tm.x; aM.y = tm.y;
        aX.x = tx.x; aX.y = tx.y;
        bL.x = tl.x; bL.y = tl.y;
        bR.x = tr.x; bR.y = tr.y;

        acc = __builtin_amdgcn_wmma_f32_16x16x4_f32(false, aM, false, bL,
                                                    (short)0, acc, false, false);
        acc = __builtin_amdgcn_wmma_f32_16x16x4_f32(false, aX, false, bR,
                                                    (short)0, acc, false, false);
    }

    const float bias = bl[col];
#pragma unroll
    for (int j = 0; j < 8; ++j) {
        const int r = rowBase + j + 8 * half;
        float v = acc[j] + bias;
        if (do_relu) v = fmaxf(v, 0.0f);
        out[(size_t)r * HID + col] = v;
    }
}

// ---------------------------------------------------------------------------
// Launch: zero -> scatter(x)+count -> gemm1(relu) -> zero -> scatter(h) -> gemm2
// ---------------------------------------------------------------------------
extern "C" void kernel_launch(void* const* d_in, const int* in_sizes, int n_in,
                              void* d_out, int out_size, void* d_ws, size_t ws_size,
                              hipStream_t stream) {
    const float*     x   = (const float*)d_in[0];
    const long long* ei  = (const long long*)d_in[1];   // int64 [2, E]
    const float*     W1l = (const float*)d_in[2];
    const float*     b1l = (const float*)d_in[3];
    const float*     W1r = (const float*)d_in[4];
    const float*     W2l = (const float*)d_in[5];
    const float*     b2l = (const float*)d_in[6];
    const float*     W2r = (const float*)d_in[7];
    float*           out = (float*)d_out;

    float* ws  = (float*)d_ws;
    float* agg = ws;                         // [N,128]
    float* cnt = ws + (size_t)NN * HID;      // [N]
    float* h   = cnt + NN;                   // [N,128]

    const long long aggCntElems = (long long)NN * HID + NN;
    const int zgrid = 2048, zblk = 256;
    const int sgrid = (EE + 7) / 8;          // 1 wave per edge, 8 edges/block
    const int ggrid = NN / 16;               // 3125 row tiles

    // Layer 1
    zero_kernel<<<zgrid, zblk, 0, stream>>>(agg, aggCntElems);
    scatter_kernel<<<sgrid, 256, 0, stream>>>(x, ei, agg, cnt, EE);
    sage_gemm_kernel<<<ggrid, 256, 0, stream>>>(agg, cnt, x, W1l, b1l, W1r, h, 1);

    // Layer 2 (reuse agg, keep cnt)
    zero_kernel<<<zgrid, zblk, 0, stream>>>(agg, (long long)NN * HID);
    scatter_kernel<<<sgrid, 256, 0, stream>>>(h, ei, agg, nullptr, EE);
    sage_gemm_kernel<<<ggrid, 256, 0, stream>>>(agg, cnt, h, W2l, b2l, W2r, out, 0);
}